// rGenerator_41455024341155
// MI455X (gfx1250) — compile-verified
//
#include <hip/hip_runtime.h>
#include <hip/hip_bf16.h>
#include <stdint.h>

// Problem constants from the reference (B,T,NF,NC,H = 4096,4096,1,10,1)
#define Bn 4096
#define Tn 4096
#define CT 32                 // timesteps staged per chunk
#define BLK 128               // threads per block = sequences per block (4 waves)
#define NCHUNK (Tn / CT)      // 128
#define ROWF 68               // floats per seq row in LDS (32 t * 2 ch + 4 pad) = 272B, 16B aligned
#define BUFF (BLK * ROWF)     // floats per LDS buffer

// --- CDNA5-specific primitives -------------------------------------------

// v_tanh_f32: gfx1250 transcendental. v_nop after it satisfies the TRANS-op
// co-execution hazard (ISA 7.4: 1 independent op / V_NOP before result use),
// since inline asm bypasses compiler hazard scheduling.
__device__ __forceinline__ float tanh_fast(float x) {
  float r;
  asm volatile("v_tanh_f32 %0, %1\n\tv_nop" : "=v"(r) : "v"(x));
  return r;
}

// Async global->LDS copy (CDNA5 async path, tracked by ASYNCcnt).
// dst operand is the LDS byte address in a VGPR, src is a 64-bit global addr.
__device__ __forceinline__ void async_load_b128(unsigned lds, const void* g) {
  asm volatile("global_load_async_to_lds_b128 %0, %1, off"
               :: "v"(lds), "v"(g) : "memory");
}
__device__ __forceinline__ void wait_async0() {
  asm volatile("s_wait_asynccnt 0" ::: "memory");
}

// -------------------------------------------------------------------------

__global__ __launch_bounds__(BLK) void gru_h1_kernel(
    const float* __restrict__ xi, const float* __restrict__ labels,
    const float* __restrict__ Wih, const float* __restrict__ Whh,
    const float* __restrict__ bih, const float* __restrict__ bhh,
    float* __restrict__ out)
{
  __shared__ __align__(16) float sIn[2 * BUFF];

  const int j  = threadIdx.x;
  const int b0 = blockIdx.x * BLK;
  const int b  = b0 + j;
  const int sj = j >> 4;      // sequence row handled by this lane during staging
  const int pj = j & 15;      // 16B segment within that row's chunk (16 segs = 256B)

  // Weights: Wih is (3,2) row-major, gate order r,z,n.
  // Pre-halve the r/z gate constants so sigmoid(x) = 0.5*tanh(halved-arg)+0.5
  // needs no extra multiply on the critical path.
  const float lab  = labels[(size_t)b * Tn];
  const float wr0h = 0.5f * Wih[0];
  const float wz0h = 0.5f * Wih[2];
  const float wn0  = Wih[4];
  const float cirh = fmaf(lab, 0.5f * Wih[1], 0.5f * bih[0]);
  const float cizh = fmaf(lab, 0.5f * Wih[3], 0.5f * bih[1]);
  const float cin  = fmaf(lab, Wih[5], bih[2]);
  const float whrh = 0.5f * Whh[0], whzh = 0.5f * Whh[1], whnh = 0.5f * Whh[2];
  const float bhrh = 0.5f * bhh[0], bhzh = 0.5f * bhh[1], bhnh = 0.5f * bhh[2];

  // LDS flat address: low 32 bits of the generic pointer are the LDS offset.
  const unsigned ldsBase  = (unsigned)(uintptr_t)(&sIn[0]);
  const size_t   rowbytes = (size_t)Tn * 8;   // bytes per sequence in xi (2 f32/t)
  // Staging bases for this thread: sequence (b0+sj), 16B segment pj.
  const char*    gthread  = (const char*)xi + (size_t)(b0 + sj) * rowbytes
                                            + (size_t)pj * 16;
  const unsigned lthread  = ldsBase + (unsigned)(sj * (ROWF * 4) + pj * 16);

  auto stage = [&](int buf, int c) {
    const char* g = gthread + (size_t)c * (CT * 8);
    unsigned    l = lthread + (unsigned)buf * (BUFF * 4);
#pragma unroll
    for (int k = 0; k < 16; ++k)   // 16 async B128s, rows 8 sequences apart
      async_load_b128(l + k * (8 * ROWF * 4), g + (size_t)k * 8 * rowbytes);
  };

  stage(0, 0);
  wait_async0();
  __syncthreads();

  float h = 0.0f;
  float* outp = out + (size_t)b * Tn;

  for (int c = 0; c < NCHUNK; ++c) {
    const int cur = c & 1;
    if (c + 1 < NCHUNK) stage(cur ^ 1, c + 1);   // prefetch next chunk

    const float* row = &sIn[cur * BUFF + j * ROWF];
    float4* ostore = reinterpret_cast<float4*>(outp + c * CT);
#pragma unroll
    for (int t4 = 0; t4 < CT / 4; ++t4) {
      float4 ov;
      float o[4];
#pragma unroll
      for (int u = 0; u < 4; ++u) {
        const int  tt  = t4 * 4 + u;
        const float x0 = row[tt * 2];            // channel 0 only
        // halved input-gate args (parallel, off h-chain except via fma latency)
        const float ir2 = fmaf(x0, wr0h, cirh);
        const float iz2 = fmaf(x0, wz0h, cizh);
        const float inn = fmaf(x0, wn0, cin);
        // halved hidden-gate args (depend on h)
        const float hr2   = fmaf(h, whrh, bhrh);
        const float hz2   = fmaf(h, whzh, bhzh);
        const float hn2   = fmaf(h, whnh, bhnh); // = 0.5*hn
        const float hhalf = 0.5f * h;
        const float tr = tanh_fast(ir2 + hr2);   // r = 0.5*tr + 0.5
        const float tz = tanh_fast(iz2 + hz2);   // z = 0.5*tz + 0.5
        // n = tanh(inn + r*hn) = tanh(inn + hn2 + tr*hn2)
        const float n  = tanh_fast(fmaf(tr, hn2, inn + hn2));
        // h' = (1-z)*n + z*h = (0.5h+0.5n) + tz*(0.5h-0.5n)
        const float a  = fmaf(-0.5f, n, hhalf);
        const float bb = fmaf( 0.5f, n, hhalf);
        h = fmaf(tz, a, bb);
        o[u] = tanh_fast(h);                     // final elementwise tanh fused
      }
      ov.x = o[0]; ov.y = o[1]; ov.z = o[2]; ov.w = o[3];
      ostore[t4] = ov;                           // one 16B store per 4 steps
    }

    wait_async0();     // next chunk's async loads (and nothing else) complete
    __syncthreads();   // make them visible to all waves in the block
  }
}

extern "C" void kernel_launch(void* const* d_in, const int* in_sizes, int n_in,
                              void* d_out, int out_size, void* d_ws, size_t ws_size,
                              hipStream_t stream) {
  const float* xi     = (const float*)d_in[0];
  const float* labels = (const float*)d_in[1];
  const float* Wih    = (const float*)d_in[2];
  const float* Whh    = (const float*)d_in[3];
  const float* bih    = (const float*)d_in[4];
  const float* bhh    = (const float*)d_in[5];
  float* out = (float*)d_out;

  const size_t BT = (size_t)Bn * Tn;

  // Output tuple = (tanh(gru(x)), labels) flattened: recurrence fills [0,BT),
  // labels are a straight d2d copy into [BT, 2*BT).
  gru_h1_kernel<<<Bn / BLK, BLK, 0, stream>>>(xi, labels, Wih, Whh, bih, bhh, out);
  hipMemcpyAsync(out + BT, labels, BT * sizeof(float),
                 hipMemcpyDeviceToDevice, stream);
}